// MyNormScanSali_18021682774593
// MI455X (gfx1250) — compile-verified
//
#include <hip/hip_runtime.h>

// -----------------------------------------------------------------------------
// MI455X reasoning:
//   Total traffic = 256*65536*(4B fp32 + 4B int32) = 134 MB, read exactly once.
//   At 23.3 TB/s HBM this is ~5.8 us; arithmetic is ~3 flops/elem -> pure
//   bandwidth problem. Strategy: b128 vector loads with non-temporal hints
//   (single-touch stream, don't pollute the 192MB L2), 2048 workgroups for MLP,
//   and deterministic hierarchical reductions. Cross-lane (wave32) reductions
//   use V_WMMA_F32_16X16X4_F32 with a ones B-matrix: D[m,n] = v(m) + v(m+16),
//   so sum(d[0..7]) + shfl_xor(16) gives the full 32-lane sum.
// -----------------------------------------------------------------------------

typedef __attribute__((ext_vector_type(2))) float v2f;
typedef __attribute__((ext_vector_type(8))) float v8f;
typedef __attribute__((ext_vector_type(4))) float f32x4;
typedef __attribute__((ext_vector_type(4))) int   i32x4;

#define BATCH      256
#define ROW_N      65536       // 256*256 elements per batch row
#define CHUNKS     8           // blocks per row
#define CHUNK_N    (ROW_N / CHUNKS)      // 8192 elements per block
#define BLOCK_T    256         // 8 wave32s
#define VEC_ITERS  (CHUNK_N / 4 / BLOCK_T)   // 8 float4 loads per thread

// Full wave32 sum via one V_WMMA_F32_16X16X4_F32.
// A(16x4): lane m VGPR0 = A[m,0] (= v), VGPR1 = A[m,1] (= 0);
//          lane m+16 VGPR0 = A[m,2], VGPR1 = A[m,3].
// B(4x16) = ones  ->  D[m,n] = v(m) + v(m+16) for every column n.
// Lanes 0-15 hold D rows 0-7 in d[0..7], lanes 16-31 hold rows 8-15, so
// sum(d[0..7]) gives half-sums; xor-16 swap combines the halves.
__device__ __forceinline__ float wave_sum32(float v) {
    v2f a; a[0] = v;    a[1] = 0.0f;
    v2f b; b[0] = 1.0f; b[1] = 1.0f;
    v8f c = {};
    v8f d = __builtin_amdgcn_wmma_f32_16x16x4_f32(
        /*neg_a=*/false, a, /*neg_b=*/false, b,
        /*c_mod=*/(short)0, c, /*reuse_a=*/false, /*reuse_b=*/false);
    float t = ((d[0] + d[1]) + (d[2] + d[3])) + ((d[4] + d[5]) + (d[6] + d[7]));
    t += __shfl_xor(t, 16, 32);
    return t;   // wave total, broadcast to all lanes
}

// Kernel 1: per-(row,chunk) partial sums {Sx, Sxx, Sxt, St} -> d_ws[block*4]
__global__ void nss_partial_kernel(const float* __restrict__ inp,
                                   const int* __restrict__ tgt,
                                   float* __restrict__ ws) {
    const int blk   = blockIdx.x;             // row * CHUNKS + chunk
    const long base = (long)blk * CHUNK_N;    // flat element offset
    const f32x4* __restrict__ in4 = (const f32x4*)(inp + base);
    const i32x4* __restrict__ tg4 = (const i32x4*)(tgt + base);

    float sx = 0.0f, sxx = 0.0f, sxt = 0.0f, st = 0.0f;
#pragma unroll
    for (int it = 0; it < VEC_ITERS; ++it) {
        const int i = it * BLOCK_T + threadIdx.x;
        f32x4 x = __builtin_nontemporal_load(&in4[i]);   // global_load_b128, TH=NT
        i32x4 t = __builtin_nontemporal_load(&tg4[i]);
#pragma unroll
        for (int k = 0; k < 4; ++k) {
            const float xv = x[k];
            const float m  = (t[k] != 0) ? 1.0f : 0.0f;
            sx  += xv;
            sxx  = fmaf(xv, xv, sxx);
            sxt  = fmaf(xv, m,  sxt);
            st  += m;
        }
    }

    // wave32 reductions via WMMA (EXEC is all-ones here: no divergence)
    const float wsx  = wave_sum32(sx);
    const float wsxx = wave_sum32(sxx);
    const float wsxt = wave_sum32(sxt);
    const float wst  = wave_sum32(st);

    __shared__ float lred[BLOCK_T / 32][4];
    const int lane = threadIdx.x & 31;
    const int wv   = threadIdx.x >> 5;
    if (lane == 0) {
        lred[wv][0] = wsx;  lred[wv][1] = wsxx;
        lred[wv][2] = wsxt; lred[wv][3] = wst;
    }
    __syncthreads();
    if (threadIdx.x < 4) {               // threads 0..3 each own one quantity
        float s = 0.0f;
#pragma unroll
        for (int w = 0; w < BLOCK_T / 32; ++w) s += lred[w][threadIdx.x];
        ws[blk * 4 + threadIdx.x] = s;
    }
}

// Kernel 2: combine chunks per row, compute nss_b, average over batch.
__global__ void nss_finalize_kernel(const float* __restrict__ ws,
                                    float* __restrict__ out) {
    const int b = threadIdx.x;            // one thread per batch row (B == 256)
    float sx = 0.0f, sxx = 0.0f, sxt = 0.0f, st = 0.0f;
#pragma unroll
    for (int s = 0; s < CHUNKS; ++s) {
        const float* p = ws + (b * CHUNKS + s) * 4;
        sx += p[0]; sxx += p[1]; sxt += p[2]; st += p[3];
    }
    const float Nf   = (float)ROW_N;
    const float mean = sx / Nf;
    const float var  = (sxx - sx * mean) / (Nf - 1.0f);   // unbiased (ddof=1)
    const float istd = rsqrtf(var);
    const float nss  = -(sxt - mean * st) * istd / Nf;

    const float wsum = wave_sum32(nss);
    __shared__ float l[BLOCK_T / 32];
    if ((threadIdx.x & 31) == 0) l[threadIdx.x >> 5] = wsum;
    __syncthreads();
    if (threadIdx.x == 0) {
        float tot = 0.0f;
#pragma unroll
        for (int w = 0; w < BLOCK_T / 32; ++w) tot += l[w];
        out[0] = tot / (float)BATCH;
    }
}

extern "C" void kernel_launch(void* const* d_in, const int* in_sizes, int n_in,
                              void* d_out, int out_size, void* d_ws, size_t ws_size,
                              hipStream_t stream) {
    const float* inp = (const float*)d_in[0];
    const int*   tgt = (const int*)d_in[1];
    float* out = (float*)d_out;
    float* ws  = (float*)d_ws;      // needs BATCH*CHUNKS*4 floats = 32 KB

    nss_partial_kernel<<<BATCH * CHUNKS, BLOCK_T, 0, stream>>>(inp, tgt, ws);
    nss_finalize_kernel<<<1, BLOCK_T, 0, stream>>>(ws, out);
}